// FMICLLoss_79663053406794
// MI455X (gfx1250) — compile-verified
//
#include <hip/hip_runtime.h>
#include <hip/hip_bf16.h>
#include <math.h>
#include <stdint.h>

// FMICL loss, fused for MI455X (gfx1250, wave32).
//   Pass 1: row-normalize z1/z2, store normalized z1 (fp32, 8MB -> L2 resident),
//           per-row sq[i] = sum(z1n^2), and accumulate positive-pair term.
//   Pass 2: symmetric Gram via V_WMMA_F32_16X16X4_F32; 32x32 tile per wave,
//           b128 fragment loads feeding 2 WMMAs each via K-slot relabeling,
//           8 independent accumulators (lo/hi split), 32-bit GVS addressing +
//           fully unrolled K loop to minimize register pressure and let all
//           4 fragment loads issue back-to-back.
//   Pass 3: combine scalar.

#define NROWS 8192
#define DIMS  256

static constexpr float ALPHA    = 1.5f;
static constexpr float EPSV     = 1e-8f;
static constexpr float NORM_EPS = 1e-12f;
// inv2s2 = 1/(2*sigma^2) with sigma = 1 -> 0.5

typedef float v2f __attribute__((ext_vector_type(2)));
typedef float v4f __attribute__((ext_vector_type(4)));
typedef float v8f __attribute__((ext_vector_type(8)));

__device__ __forceinline__ float waveReduceSum(float v) {
  #pragma unroll
  for (int off = 16; off > 0; off >>= 1)
    v += __shfl_xor(v, off, 32);
  return v;
}

__global__ void fmicl_init(double* __restrict__ acc) {
  acc[0] = 0.0;  // sum of s_pos
  acc[1] = 0.0;  // sum of off-diagonal star_neg
}

// One wave per row; 1024 blocks x 256 threads (8 waves).
__global__ void __launch_bounds__(256)
fmicl_normalize(const float* __restrict__ z1, const float* __restrict__ z2,
                float* __restrict__ zn, float* __restrict__ sq,
                double* __restrict__ acc) {
  const int wave = threadIdx.x >> 5;
  const int lane = threadIdx.x & 31;
  const int row  = blockIdx.x * 8 + wave;
  const uint32_t base = (uint32_t)row * DIMS + (uint32_t)lane * 8;

  float a[8], b[8];
  *(v4f*)&a[0] = *(const v4f*)(z1 + base);
  *(v4f*)&a[4] = *(const v4f*)(z1 + base + 4);
  *(v4f*)&b[0] = *(const v4f*)(z2 + base);
  *(v4f*)&b[4] = *(const v4f*)(z2 + base + 4);

  float s1 = 0.f, s2 = 0.f;
  #pragma unroll
  for (int i = 0; i < 8; ++i) { s1 += a[i] * a[i]; s2 += b[i] * b[i]; }
  s1 = waveReduceSum(s1);
  s2 = waveReduceSum(s2);

  const float inv1 = 1.0f / fmaxf(sqrtf(s1), NORM_EPS);
  const float inv2 = 1.0f / fmaxf(sqrtf(s2), NORM_EPS);

  float sql = 0.f, dpos = 0.f;
  #pragma unroll
  for (int i = 0; i < 8; ++i) {
    const float x = a[i] * inv1;
    const float y = b[i] * inv2;
    a[i] = x;
    sql += x * x;
    const float d = x - y;
    dpos += d * d;
  }
  sql  = waveReduceSum(sql);
  dpos = waveReduceSum(dpos);

  *(v4f*)(zn + base)     = *(const v4f*)&a[0];
  *(v4f*)(zn + base + 4) = *(const v4f*)&a[4];

  if (lane == 0) {
    sq[row] = sql;
    const float s_pos = logf(expf(-0.5f * dpos) + EPSV) + 1.0f;
    atomicAdd(&acc[0], (double)s_pos);
  }
}

// star-sum over one 16x16 WMMA accumulator.
__device__ __forceinline__ float tileStarSum(const v8f& c, int i0, int j,
                                             const float* __restrict__ sq,
                                             float sqj) {
  float s = 0.f;
  #pragma unroll
  for (int r = 0; r < 8; ++r) {
    const int i = i0 + r;
    const float d2   = fmaxf(sq[i] + sqj - 2.0f * c[r], 0.0f);
    const float star = expf(-0.5f * d2) + EPSV;     // == exp(log(g+eps)+1-1)
    s += (i == j) ? 0.0f : star;
  }
  return s;
}

// 32x32 output tile per wave; triangular tile set (ti <= tj).
// 8192 blocks x 256 threads (8 waves) cover 256x256 tiles.
//
// Fragment trick: WMMA's K-slot -> k assignment is arbitrary as long as A and
// B agree. We load b128 at column (k + 4*(lane>>4)): lane half 0 holds
// k..k+3, half 1 holds k+4..k+7. The .lo pairs feed one WMMA (contracting
// {k,k+1,k+4,k+5}), the .hi pairs feed a second ({k+2,k+3,k+6,k+7}) --
// together exactly [k, k+8) once, with half the VMEM instructions of b64.
// .lo and .hi streams accumulate into separate registers so all 8 WMMAs per
// iteration are independent (no accumulator RAW within an iteration).
__global__ void __launch_bounds__(256)
fmicl_gram(const float* __restrict__ zn, const float* __restrict__ sq,
           double* __restrict__ acc) {
  const int wave = threadIdx.x >> 5;
  const int lane = threadIdx.x & 31;
  const int w    = blockIdx.x * 8 + wave;
  const int ti   = w >> 8;          // 256 tiles per dim
  const int tj   = w & 255;
  if (ti > tj) return;              // wave-uniform: EXEC stays all-ones

  const int r0 = ti * 32, c0 = tj * 32;
  const int m  = lane & 15;
  const int kq = (lane >> 4) * 4;   // b128 K sub-offset per lane half

  // 32-bit offsets against the uniform base pointer -> GVS addressing
  // (SGPR64 base + VGPR32 offset), 1 VGPR per stream instead of 2.
  const uint32_t oa0 = (uint32_t)(r0 + m)      * DIMS + kq;
  const uint32_t oa1 = (uint32_t)(r0 + 16 + m) * DIMS + kq;
  const uint32_t ob0 = (uint32_t)(c0 + m)      * DIMS + kq;
  const uint32_t ob1 = (uint32_t)(c0 + 16 + m) * DIMS + kq;

  v8f c00a = {}, c01a = {}, c10a = {}, c11a = {};
  v8f c00b = {}, c01b = {}, c10b = {}, c11b = {};
  #pragma unroll
  for (int k = 0; k < DIMS; k += 8) {
    const v4f A0 = *(const v4f*)(zn + oa0 + k);
    const v4f A1 = *(const v4f*)(zn + oa1 + k);
    const v4f B0 = *(const v4f*)(zn + ob0 + k);
    const v4f B1 = *(const v4f*)(zn + ob1 + k);
    c00a = __builtin_amdgcn_wmma_f32_16x16x4_f32(false, A0.lo, false, B0.lo, (short)0, c00a, false, false);
    c01a = __builtin_amdgcn_wmma_f32_16x16x4_f32(false, A0.lo, false, B1.lo, (short)0, c01a, false, false);
    c10a = __builtin_amdgcn_wmma_f32_16x16x4_f32(false, A1.lo, false, B0.lo, (short)0, c10a, false, false);
    c11a = __builtin_amdgcn_wmma_f32_16x16x4_f32(false, A1.lo, false, B1.lo, (short)0, c11a, false, false);
    c00b = __builtin_amdgcn_wmma_f32_16x16x4_f32(false, A0.hi, false, B0.hi, (short)0, c00b, false, false);
    c01b = __builtin_amdgcn_wmma_f32_16x16x4_f32(false, A0.hi, false, B1.hi, (short)0, c01b, false, false);
    c10b = __builtin_amdgcn_wmma_f32_16x16x4_f32(false, A1.hi, false, B0.hi, (short)0, c10b, false, false);
    c11b = __builtin_amdgcn_wmma_f32_16x16x4_f32(false, A1.hi, false, B1.hi, (short)0, c11b, false, false);
  }
  const v8f c00 = c00a + c00b;
  const v8f c01 = c01a + c01b;
  const v8f c10 = c10a + c10b;
  const v8f c11 = c11a + c11b;

  // C/D layout: c[r] = G[i0 + r][j], i0 = rowBase + 8*(lane>>4), j = colBase + (lane&15)
  const int iband = (lane >> 4) * 8;
  const int j0 = c0 + m, j1 = c0 + 16 + m;
  const float sqj0 = sq[j0], sqj1 = sq[j1];

  float part = 0.f;
  part += tileStarSum(c00, r0 + iband,      j0, sq, sqj0);
  part += tileStarSum(c01, r0 + iband,      j1, sq, sqj1);
  part += tileStarSum(c10, r0 + 16 + iband, j0, sq, sqj0);
  part += tileStarSum(c11, r0 + 16 + iband, j1, sq, sqj1);
  part *= (ti == tj) ? 1.0f : 2.0f;   // symmetry weight

  const float tot = waveReduceSum(part);
  if (lane == 0) atomicAdd(&acc[1], (double)tot);
}

__global__ void fmicl_final(const double* __restrict__ acc,
                            float* __restrict__ out) {
  const double n = (double)NROWS;
  const double mean_pos  = acc[0] / n;
  const double mean_star = acc[1] / (n * (n - 1.0));
  out[0] = (float)(-mean_pos + (double)ALPHA * mean_star);
}

extern "C" void kernel_launch(void* const* d_in, const int* in_sizes, int n_in,
                              void* d_out, int out_size, void* d_ws, size_t ws_size,
                              hipStream_t stream) {
  const float* z1 = (const float*)d_in[0];
  const float* z2 = (const float*)d_in[1];
  float* out = (float*)d_out;

  // Workspace layout: [2 x double acc][N x float sq][N*D x float zn]
  double* acc = (double*)d_ws;
  float*  sq  = (float*)(acc + 2);
  float*  zn  = sq + NROWS;

  fmicl_init<<<1, 1, 0, stream>>>(acc);
  fmicl_normalize<<<NROWS / 8, 256, 0, stream>>>(z1, z2, zn, sq, acc);
  fmicl_gram<<<(256 * 256) / 8, 256, 0, stream>>>(zn, sq, acc);
  fmicl_final<<<1, 1, 0, stream>>>(acc, out);
}